// Soft_DTW_57793079935466
// MI455X (gfx1250) — compile-verified
//
#include <hip/hip_runtime.h>

#define BIGV 1.0e10f
#define NSEQ 256
#define DFEAT 32

typedef __attribute__((ext_vector_type(2))) float v2f;
typedef __attribute__((ext_vector_type(8))) float v8f;

// One workgroup (256 threads = 8 wave32) per batch element.
// Phase 0: squared norms of x,y rows -> LDS
// Phase 1: D = |x|^2 + |y|^2 - 2 x.y^T via V_WMMA_F32_16X16X4_F32 -> LDS (256 KB)
// Phase 2: anti-diagonal soft-DTW wavefront, one row per thread,
//          3 rotating diagonal buffers in LDS, one barrier per step.
__global__ __launch_bounds__(256, 1)
void sdtw_fused_kernel(const float* __restrict__ x,
                       const float* __restrict__ y,
                       float* __restrict__ out) {
    extern __shared__ float smem[];
    float* Dmat = smem;               // 256*256 floats, pitch 256 (conflict-free diagonal reads)
    float* xsq  = smem + 65536;       // 256
    float* ysq  = xsq + 256;          // 256
    float* bufA = ysq + 256;          // 257  (step s-2)
    float* bufB = bufA + 257;         // 257  (step s-1)
    float* bufC = bufB + 257;         // 257  (step s, write target)

    const int b    = blockIdx.x;
    const int tid  = threadIdx.x;
    const int lane = tid & 31;
    const int wave = tid >> 5;

    const float* xb = x + (size_t)b * NSEQ * DFEAT;
    const float* yb = y + (size_t)b * NSEQ * DFEAT;

    // ---------------- Phase 0: row norms ----------------
    {
        float sx = 0.f, sy = 0.f;
        const float* xr = xb + tid * DFEAT;
        const float* yr = yb + tid * DFEAT;
        #pragma unroll
        for (int k = 0; k < DFEAT; ++k) {
            float xv = xr[k]; sx += xv * xv;
            float yv = yr[k]; sy += yv * yv;
        }
        xsq[tid] = sx;
        ysq[tid] = sy;
    }
    // DP diagonal buffer init: step0 has R(0,0)=0, everything else BIG
    bufA[tid] = (tid == 0) ? 0.0f : BIGV;
    bufB[tid] = BIGV;
    bufC[tid] = BIGV;
    if (tid == 0) { bufA[256] = BIGV; bufB[256] = BIGV; bufC[256] = BIGV; }
    __syncthreads();

    // ---------------- Phase 1: distance matrix via fp32 WMMA ----------------
    // A fragment (16x4 f32, ISA 7.12.2): lanes 0-15 hold M=lane, K=0..1;
    // lanes 16-31 hold M=lane-16, K=2..3.  B (4x16) is the symmetric striping,
    // and B[k][n] = y[n][k], so both fragments load identically from x / y.
    const int lrow = lane & 15;
    const int lko  = (lane >> 4) * 2;
    for (int rb = wave; rb < 16; rb += 8) {          // row-tile per wave
        v2f afrag[8];
        const float* xr = xb + (rb * 16 + lrow) * DFEAT + lko;
        #pragma unroll
        for (int kk = 0; kk < 8; ++kk)
            afrag[kk] = *(const v2f*)(xr + kk * 4);

        for (int cb = 0; cb < 16; ++cb) {            // 16 column tiles
            const float* yr = yb + (cb * 16 + lrow) * DFEAT + lko;
            v8f c = {0.f, 0.f, 0.f, 0.f, 0.f, 0.f, 0.f, 0.f};
            #pragma unroll
            for (int kk = 0; kk < 8; ++kk) {         // K = 32 in steps of 4
                v2f bfrag = *(const v2f*)(yr + kk * 4);
                c = __builtin_amdgcn_wmma_f32_16x16x4_f32(
                        false, afrag[kk], false, bfrag,
                        (short)0, c, false, false);
            }
            // C/D layout: VGPR v -> M = v + 8*(lane>=16), N = lane&15
            const int mbase = rb * 16 + 8 * (lane >> 4);
            const int col   = cb * 16 + lrow;
            const float yq  = ysq[col];
            #pragma unroll
            for (int v = 0; v < 8; ++v) {
                const int row = mbase + v;
                Dmat[row * 256 + col] = xsq[row] + yq - 2.0f * c[v];
            }
        }
    }
    __syncthreads();

    // ---------------- Phase 2: soft-DTW wavefront DP ----------------
    // Thread tid owns DP row i = tid+1.  At step s it computes R(i, s-i):
    //   up   = R(i-1, j)   = neighbor's value from step s-1  (LDS)
    //   diag = R(i-1, j-1) = neighbor's value from step s-2  (LDS)
    //   left = R(i, j-1)   = own value from step s-1         (register)
    const int i = tid + 1;
    float left = BIGV;
    float* bm2 = bufA;
    float* bm1 = bufB;
    float* bw  = bufC;
    for (int s = 2; s <= 2 * NSEQ; ++s) {
        const float up   = bm1[i - 1];
        const float diag = bm2[i - 1];
        const int j = s - i;
        float val = BIGV;
        if (j >= 1 && j <= NSEQ) {
            const float d = Dmat[(i - 1) * 256 + (j - 1)];
            // softmin3 with gamma=1 (matches reference arithmetic exactly):
            //   min - log(sum exp(min - a_k))
            const float m = fminf(fminf(diag, up), left);
            const float sum = __expf(m - diag) + __expf(m - up) + __expf(m - left);
            val = d + (m - __logf(sum));
        }
        bw[i] = val;                    // writes go to a buffer nobody reads this step
        if (tid == 0) bw[0] = BIGV;     // maintain R(0, s) = BIG boundary
        left = val;
        float* t = bm2; bm2 = bm1; bm1 = bw; bw = t;
        __syncthreads();                // single barrier per wavefront step
    }
    if (tid == 255) out[b] = left;      // R(N, M) produced at s = N+M by row 256
}

extern "C" void kernel_launch(void* const* d_in, const int* in_sizes, int n_in,
                              void* d_out, int out_size, void* d_ws, size_t ws_size,
                              hipStream_t stream) {
    (void)n_in; (void)out_size; (void)d_ws; (void)ws_size;
    const float* x = (const float*)d_in[0];
    const float* y = (const float*)d_in[1];
    float* out = (float*)d_out;

    const int B = in_sizes[0] / (NSEQ * DFEAT);   // 512
    const size_t shmem = (size_t)(65536 + 256 + 256 + 3 * 257) * sizeof(float); // ~261 KB

    // Opt in to >64KB dynamic LDS (WGP has 320 KB on CDNA5). Idempotent, deterministic.
    hipFuncSetAttribute(reinterpret_cast<const void*>(sdtw_fused_kernel),
                        hipFuncAttributeMaxDynamicSharedMemorySize, (int)shmem);

    sdtw_fused_kernel<<<B, 256, shmem, stream>>>(x, y, out);
}